// RGCN_1778116460979
// MI455X (gfx1250) — compile-verified
//
#include <hip/hip_runtime.h>

typedef float v2f __attribute__((ext_vector_type(2)));
typedef float v8f __attribute__((ext_vector_type(8)));

#define D 128
#define XS_STRIDE 130   // pad to kill 64-bank conflicts (130 mod 64 = 2)
#define WT_STRIDE 130

// ---------------- utility kernels ----------------

__global__ void zero_kernel(float* __restrict__ p, int n) {
    int i = blockIdx.x * blockDim.x + threadIdx.x;
    if (i < n) p[i] = 0.0f;
}

__global__ void degree_kernel(const int* __restrict__ src, const int* __restrict__ dst,
                              float* __restrict__ dout, float* __restrict__ din, int e) {
    int i = blockIdx.x * blockDim.x + threadIdx.x;
    if (i < e) {
        atomicAdd(&dout[src[i]], 1.0f);
        atomicAdd(&din[dst[i]], 1.0f);
    }
}

__global__ void rsqrt_kernel(float* __restrict__ deg, int n) {
    int i = blockIdx.x * blockDim.x + threadIdx.x;
    if (i < n) deg[i] = rsqrtf(fmaxf(deg[i], 1.0f));
}

// out = relu(out + b0 + b1 + b2)
__global__ void finish_kernel(float* __restrict__ out, const float* __restrict__ b0,
                              const float* __restrict__ b1, const float* __restrict__ b2,
                              int total) {
    int i = blockIdx.x * blockDim.x + threadIdx.x;
    if (i < total) {
        int j = i & (D - 1);
        float v = out[i] + b0[j] + b1[j] + b2[j];
        out[i] = v > 0.0f ? v : 0.0f;
    }
}

// ---------------- GEMM: m = (x * sout[:,None]) @ W  via V_WMMA_F32_16X16X4_F32 ----------------
// One block = 16 output rows x 128 output cols. 8 waves, each wave owns a 16x16 tile.
// K loop: 32 steps of K=4.

__global__ __launch_bounds__(256) void gemm_kernel(const float* __restrict__ x,
                                                   const float* __restrict__ W,
                                                   const float* __restrict__ sout,
                                                   float* __restrict__ m, int n) {
    __shared__ float xs[16 * XS_STRIDE];     // scaled A tile, row-major [row][k]
    __shared__ float wt[D * WT_STRIDE];      // W transposed: wt[ncol][k]

    const int row0 = blockIdx.x * 16;
    const int tid  = threadIdx.x;
    if (row0 >= n) return;

    // stage x tile (16x128), pre-scaled by sout[row]; 512 float4s across 256 threads
    for (int i = tid; i < 16 * (D / 4); i += 256) {
        int r  = i >> 5;          // /(D/4)
        int c4 = i & 31;
        float4 v = ((const float4*)(x + (size_t)(row0 + r) * D))[c4];
        float  s = sout[row0 + r];
        float* p = &xs[r * XS_STRIDE + c4 * 4];
        p[0] = v.x * s; p[1] = v.y * s; p[2] = v.z * s; p[3] = v.w * s;
    }
    // stage W transposed: W row-major [k][ncol] -> wt[ncol][k]
    for (int i = tid; i < (D * D) / 4; i += 256) {
        int k  = i >> 5;
        int c4 = i & 31;
        float4 v = ((const float4*)(W + (size_t)k * D))[c4];
        int n0 = c4 * 4;
        wt[(n0 + 0) * WT_STRIDE + k] = v.x;
        wt[(n0 + 1) * WT_STRIDE + k] = v.y;
        wt[(n0 + 2) * WT_STRIDE + k] = v.z;
        wt[(n0 + 3) * WT_STRIDE + k] = v.w;
    }
    __syncthreads();

    const int wave  = tid >> 5;          // 0..7 -> 16-col tile
    const int lane  = tid & 31;
    const int col0  = wave * 16;
    const int lrow  = lane & 15;
    const int khalf = (lane >> 4) * 2;   // lanes 0-15: K offset 0, lanes 16-31: K offset 2

    // A 16x4 f32 layout: VGPR0 = K{0|2}, VGPR1 = K{1|3}; B mirrored (4x16).
    const float* ap = &xs[lrow * XS_STRIDE + khalf];
    const float* bp = &wt[(col0 + lrow) * WT_STRIDE + khalf];

    v8f c = {};
#pragma unroll
    for (int k0 = 0; k0 < D; k0 += 4) {
        v2f a = *(const v2f*)(ap + k0);   // 8B-aligned ds_load_b64
        v2f b = *(const v2f*)(bp + k0);
        c = __builtin_amdgcn_wmma_f32_16x16x4_f32(
                /*neg_a=*/false, a, /*neg_b=*/false, b,
                /*c_mod=*/(short)0, c, /*reuse_a=*/false, /*reuse_b=*/false);
    }

    // C/D layout: VGPR v -> row = v + (lane>=16 ? 8 : 0), col = lane&15
    const int crow = row0 + (lane >> 4) * 8;
    const int ccol = col0 + (lane & 15);
#pragma unroll
    for (int v = 0; v < 8; ++v) {
        m[(size_t)(crow + v) * D + ccol] = c[v];
    }
}

// ---------------- scatter: out[dst] += m[src] * sin[dst], wave per edge ----------------

__global__ __launch_bounds__(256) void scatter_kernel(const float* __restrict__ m,
                                                      const int* __restrict__ src,
                                                      const int* __restrict__ dst,
                                                      const float* __restrict__ sin_,
                                                      float* __restrict__ out, int e) {
    int gid  = blockIdx.x * blockDim.x + threadIdx.x;
    int edge = gid >> 5;
    int lane = threadIdx.x & 31;
    if (edge >= e) return;
    int s = src[edge];
    int d = dst[edge];
    float sc = sin_[d];                                  // uniform across wave
    float4 v = ((const float4*)(m + (size_t)s * D))[lane]; // 512B coalesced per wave
    float* o = out + (size_t)d * D + lane * 4;
    atomicAdd(o + 0, v.x * sc);
    atomicAdd(o + 1, v.y * sc);
    atomicAdd(o + 2, v.z * sc);
    atomicAdd(o + 3, v.w * sc);
}

// ---------------- launcher ----------------

extern "C" void kernel_launch(void* const* d_in, const int* in_sizes, int n_in,
                              void* d_out, int out_size, void* d_ws, size_t ws_size,
                              hipStream_t stream) {
    // dict order: x, (W0,b0,src0,dst0), (W1,b1,src1,dst1), (W2,b2,src2,dst2)
    const float* x = (const float*)d_in[0];
    const float* W[3]   = {(const float*)d_in[1], (const float*)d_in[5], (const float*)d_in[9]};
    const float* b[3]   = {(const float*)d_in[2], (const float*)d_in[6], (const float*)d_in[10]};
    const int*   src[3] = {(const int*)d_in[3],   (const int*)d_in[7],   (const int*)d_in[11]};
    const int*   dst[3] = {(const int*)d_in[4],   (const int*)d_in[8],   (const int*)d_in[12]};
    float* out = (float*)d_out;

    const int Nn = in_sizes[0] / D;   // 100000
    const int E  = in_sizes[3];       // 600000

    // workspace: m[N*D] | degs[6*N]  (sout0,sin0,sout1,sin1,sout2,sin2)
    float* m    = (float*)d_ws;
    float* degs = m + (size_t)Nn * D;

    // 1) zero output accumulator + degree buffers
    zero_kernel<<<(Nn * D + 255) / 256, 256, 0, stream>>>(out, Nn * D);
    zero_kernel<<<(6 * Nn + 255) / 256, 256, 0, stream>>>(degs, 6 * Nn);

    // 2) degrees per relation
    for (int r = 0; r < 3; ++r) {
        float* dout_r = degs + (size_t)r * 2 * Nn;
        float* din_r  = dout_r + Nn;
        degree_kernel<<<(E + 255) / 256, 256, 0, stream>>>(src[r], dst[r], dout_r, din_r, E);
    }

    // 3) deg -> rsqrt(max(deg,1)) in place
    rsqrt_kernel<<<(6 * Nn + 255) / 256, 256, 0, stream>>>(degs, 6 * Nn);

    // 4) per relation: WMMA GEMM then edge scatter (stream order gives the dependency)
    for (int r = 0; r < 3; ++r) {
        float* sout_r = degs + (size_t)r * 2 * Nn;
        float* sin_r  = sout_r + Nn;
        gemm_kernel<<<(Nn + 15) / 16, 256, 0, stream>>>(x, W[r], sout_r, m, Nn);
        scatter_kernel<<<(E + 7) / 8, 256, 0, stream>>>(m, src[r], dst[r], sin_r, out, E);
    }

    // 5) bias + relu
    finish_kernel<<<(Nn * D + 255) / 256, 256, 0, stream>>>(out, b[0], b[1], b[2], Nn * D);
}